// ILRMA_6803228197167
// MI455X (gfx1250) — compile-verified
//
#include <hip/hip_runtime.h>
#include <math.h>

#define JD 2048
#define ID 1025
#define IPD 1040      // 65 * 16, padded frequency dim
#define KB 8
#define MC 2
#define NSRC 2
#define NITER 5
#define YCH 8         // blockIdx.y split of the reduction dim in NMF kernels
#define EPS 1e-20f    // NMF_EPS == IP_EPS == 1e-20

typedef __attribute__((ext_vector_type(2))) float v2f;
typedef __attribute__((ext_vector_type(8))) float v8f;

// ---- workspace layout (float offsets) ----
constexpr size_t oXre = 0;
constexpr size_t oXim = oXre + (size_t)MC * IPD * JD;
constexpr size_t oYre = oXim + (size_t)MC * IPD * JD;
constexpr size_t oYim = oYre + (size_t)NSRC * IPD * JD;
constexpr size_t oYdR = oYim + (size_t)NSRC * IPD * JD;
constexpr size_t oT   = oYdR + (size_t)IPD * JD;
constexpr size_t oV   = oT + (size_t)NSRC * IPD * KB;
constexpr size_t oW   = oV + (size_t)NSRC * KB * JD;   // W: [i][n][m][re/im] = i*8 + n*4 + m*2 + c
constexpr size_t oPTn = oW + (size_t)ID * 8;           // T-update partials [YCH][IPD][KB]
constexpr size_t oPTd = oPTn + (size_t)YCH * IPD * KB;
constexpr size_t oPVn = oPTd + (size_t)YCH * IPD * KB; // V-update partials [YCH][KB][JD]
constexpr size_t oPVd = oPVn + (size_t)YCH * KB * JD;

__device__ __forceinline__ v8f wmma_k4(v2f a, v2f b, v8f c) {
  // D(16x16,f32) = A(16x4,f32) * B(4x16,f32) + C
  return __builtin_amdgcn_wmma_f32_16x16x4_f32(false, a, false, b, (short)0, c, false, false);
}

// ---------------- init kernels ----------------
__global__ __launch_bounds__(256) void k_init_xy(const float* __restrict__ X, float* __restrict__ ws) {
  size_t t = (size_t)blockIdx.x * 256 + threadIdx.x;
  size_t tot = (size_t)MC * IPD * JD;
  if (t >= tot) return;
  int j = (int)(t % JD); size_t r = t / JD;
  int i = (int)(r % IPD); int m = (int)(r / IPD);
  float xr = 0.f, xi = 0.f;
  if (i < ID) {
    size_t src = (((size_t)m * JD + j) * ID + i) * 2;
    xr = X[src]; xi = X[src + 1];
  }
  size_t dst = ((size_t)m * IPD + i) * JD + j;
  ws[oXre + dst] = xr; ws[oXim + dst] = xi;
  // W0 = eye(N,M) with N==M==2  =>  Y0 = Xc
  ws[oYre + dst] = xr; ws[oYim + dst] = xi;
}

__global__ __launch_bounds__(256) void k_init_t(const float* __restrict__ T0, float* __restrict__ ws) {
  size_t t = (size_t)blockIdx.x * 256 + threadIdx.x;
  size_t tot = (size_t)NSRC * IPD * KB;
  if (t >= tot) return;
  int k = (int)(t % KB); size_t r = t / KB;
  int i = (int)(r % IPD); int n = (int)(r / IPD);
  float v = 0.f;
  if (i < ID) v = T0[((size_t)i * KB + k) * NSRC + n];
  ws[oT + ((size_t)n * IPD + i) * KB + k] = v;
}

__global__ __launch_bounds__(256) void k_init_v(const float* __restrict__ V0, float* __restrict__ ws) {
  size_t t = (size_t)blockIdx.x * 256 + threadIdx.x;
  size_t tot = (size_t)NSRC * KB * JD;
  if (t >= tot) return;
  int j = (int)(t % JD); size_t r = t / JD;
  int k = (int)(r % KB); int n = (int)(r / KB);
  ws[oV + ((size_t)n * KB + k) * JD + j] = V0[((size_t)k * JD + j) * NSRC + n];
}

__global__ __launch_bounds__(256) void k_init_w(float* __restrict__ ws) {
  size_t t = (size_t)blockIdx.x * 256 + threadIdx.x;
  size_t tot = (size_t)ID * 8;
  if (t >= tot) return;
  int f = (int)(t & 7);
  int c = f & 1, m = (f >> 1) & 1, nn = (f >> 2) & 1;
  ws[oW + t] = (c == 0 && nn == m) ? 1.f : 0.f;
}

// ---------------- NMF: T update stage 1 (per source) ----------------
// grid (IPD/16, YCH). Per 16-row i-tile: Rn = T@V via WMMA, YdR/Rd elementwise
// (YdR stored to ws), then num = YdR@V^T, den = Rd@V^T via WMMA over this
// block's slice of j; per-block partial written to oPTn/oPTd.
__global__ __launch_bounds__(256) void k_t_update(float* __restrict__ ws, int n) {
  const int tid = threadIdx.x;
  const int wv = tid >> 5, lane = tid & 31, hf = lane >> 4, l16 = lane & 15;
  const int i0 = blockIdx.x * 16;
  const int yb = blockIdx.y;
  const float* T = ws + oT + (size_t)n * IPD * KB;
  const float* V = ws + oV + (size_t)n * KB * JD;
  const float* Yre = ws + oYre + (size_t)n * IPD * JD;
  const float* Yim = ws + oYim + (size_t)n * IPD * JD;
  float* YdR = ws + oYdR;

  __shared__ float lds_y[8][16][17];
  __shared__ float lds_r[8][16][17];
  __shared__ float redn[8][256];
  __shared__ float redd[8][256];

  v2f a0, a1;
  {
    size_t tr = (size_t)(i0 + l16) * KB;
    a0.x = T[tr + 2 * hf];     a0.y = T[tr + 2 * hf + 1];
    a1.x = T[tr + 4 + 2 * hf]; a1.y = T[tr + 5 + 2 * hf];
  }

  v8f num = {0,0,0,0,0,0,0,0}, den = {0,0,0,0,0,0,0,0};

  // 128 j-tiles / (YCH*8 waves) = 2 tiles per wave
  for (int jt = yb * 8 + wv; jt < JD / 16; jt += YCH * 8) {
    const int j0 = jt * 16;
    v2f b0, b1;
    b0.x = V[(size_t)(2 * hf + 0) * JD + j0 + l16];
    b0.y = V[(size_t)(2 * hf + 1) * JD + j0 + l16];
    b1.x = V[(size_t)(2 * hf + 4) * JD + j0 + l16];
    b1.y = V[(size_t)(2 * hf + 5) * JD + j0 + l16];
    v8f c = {0,0,0,0,0,0,0,0};
    c = wmma_k4(a0, b0, c);
    c = wmma_k4(a1, b1, c);
#pragma unroll
    for (int r = 0; r < 8; r++) {
      const int il = r + 8 * hf;
      size_t idx = (size_t)(i0 + il) * JD + (j0 + l16);
      float yr = Yre[idx], yi = Yim[idx];
      float rn = c[r];
      float ydr = (yr * yr + yi * yi) / (rn * rn + EPS);
      float rd = 1.0f / (rn + EPS);
      YdR[idx] = ydr;
      lds_y[wv][il][l16] = ydr;
      lds_r[wv][il][l16] = rd;
    }
    // num/den tiles: inner dim = 16 j's of this tile, 4 chained K=4 WMMAs
#pragma unroll
    for (int jc = 0; jc < 16; jc += 4) {
      v2f ay, ar, bv;
      ay.x = lds_y[wv][l16][jc + 2 * hf]; ay.y = lds_y[wv][l16][jc + 2 * hf + 1];
      ar.x = lds_r[wv][l16][jc + 2 * hf]; ar.y = lds_r[wv][l16][jc + 2 * hf + 1];
      float bx = 0.f, by = 0.f;
      if (l16 < KB) {
        bx = V[(size_t)l16 * JD + j0 + jc + 2 * hf];
        by = V[(size_t)l16 * JD + j0 + jc + 2 * hf + 1];
      }
      bv.x = bx; bv.y = by;
      num = wmma_k4(ay, bv, num);
      den = wmma_k4(ar, bv, den);
    }
  }
#pragma unroll
  for (int r = 0; r < 8; r++) { redn[wv][r * 32 + lane] = num[r]; redd[wv][r * 32 + lane] = den[r]; }
  __syncthreads();
  if (wv == 0) {
#pragma unroll
    for (int r = 0; r < 8; r++) {
      float sn = 0.f, sd = 0.f;
      for (int w2 = 0; w2 < 8; w2++) { sn += redn[w2][r * 32 + lane]; sd += redd[w2][r * 32 + lane]; }
      if (l16 < KB) {
        int ii = i0 + r + 8 * hf;   // C layout: M = r + 8*half (i), N = l16 (k)
        ws[oPTn + ((size_t)yb * IPD + ii) * KB + l16] = sn;
        ws[oPTd + ((size_t)yb * IPD + ii) * KB + l16] = sd;
      }
    }
  }
}

// ---------------- NMF: T update stage 2 (deterministic partial sum + apply) ----------------
__global__ __launch_bounds__(256) void k_t_apply(float* __restrict__ ws, int n) {
  size_t t = (size_t)blockIdx.x * 256 + threadIdx.x;
  size_t tot = (size_t)IPD * KB;
  if (t >= tot) return;
  float sn = 0.f, sd = 0.f;
#pragma unroll
  for (int y = 0; y < YCH; y++) {
    sn += ws[oPTn + (size_t)y * IPD * KB + t];
    sd += ws[oPTd + (size_t)y * IPD * KB + t];
  }
  ws[oT + (size_t)n * IPD * KB + t] *= sqrtf(sn / (sd + EPS));
}

// ---------------- NMF: V update stage 1 (per source) ----------------
// grid (JD/16, YCH). Per 16-col j-tile: Rn = Tnew@V via WMMA -> Rd2;
// numV = T^T@YdR, denV = T^T@Rd2 via WMMA over this block's slice of i.
__global__ __launch_bounds__(256) void k_v_update(float* __restrict__ ws, int n) {
  const int tid = threadIdx.x;
  const int wv = tid >> 5, lane = tid & 31, hf = lane >> 4, l16 = lane & 15;
  const int j0 = blockIdx.x * 16;
  const int yb = blockIdx.y;
  const float* T = ws + oT + (size_t)n * IPD * KB;
  const float* V = ws + oV + (size_t)n * KB * JD;
  const float* YdR = ws + oYdR;

  __shared__ float lds_r[8][16][17];
  __shared__ float redn[8][256];
  __shared__ float redd[8][256];

  v2f b0, b1;
  b0.x = V[(size_t)(2 * hf + 0) * JD + j0 + l16];
  b0.y = V[(size_t)(2 * hf + 1) * JD + j0 + l16];
  b1.x = V[(size_t)(2 * hf + 4) * JD + j0 + l16];
  b1.y = V[(size_t)(2 * hf + 5) * JD + j0 + l16];

  v8f num = {0,0,0,0,0,0,0,0}, den = {0,0,0,0,0,0,0,0};

  // 65 i-tiles over YCH*8 = 64 global waves (first wave gets 2 tiles)
  for (int it = yb * 8 + wv; it < IPD / 16; it += YCH * 8) {
    const int i0 = it * 16;
    v2f a0, a1;
    size_t tr = (size_t)(i0 + l16) * KB;
    a0.x = T[tr + 2 * hf];     a0.y = T[tr + 2 * hf + 1];
    a1.x = T[tr + 4 + 2 * hf]; a1.y = T[tr + 5 + 2 * hf];
    v8f c = {0,0,0,0,0,0,0,0};
    c = wmma_k4(a0, b0, c);
    c = wmma_k4(a1, b1, c);
#pragma unroll
    for (int r = 0; r < 8; r++)
      lds_r[wv][r + 8 * hf][l16] = 1.0f / (fabsf(c[r]) + EPS);
    // numV/denV tiles: inner dim = 16 i's of this chunk
#pragma unroll
    for (int ic = 0; ic < 16; ic += 4) {
      v2f at, byv, brv;
      int ia = i0 + ic + 2 * hf;
      at.x = (l16 < KB) ? T[(size_t)ia * KB + l16] : 0.f;        // A = T^T: M=k, K=i
      at.y = (l16 < KB) ? T[(size_t)(ia + 1) * KB + l16] : 0.f;
      byv.x = YdR[(size_t)ia * JD + j0 + l16];                   // B = YdR: row=i, col=j
      byv.y = YdR[(size_t)(ia + 1) * JD + j0 + l16];
      brv.x = lds_r[wv][ic + 2 * hf][l16];
      brv.y = lds_r[wv][ic + 2 * hf + 1][l16];
      num = wmma_k4(at, byv, num);
      den = wmma_k4(at, brv, den);
    }
  }
#pragma unroll
  for (int r = 0; r < 8; r++) { redn[wv][r * 32 + lane] = num[r]; redd[wv][r * 32 + lane] = den[r]; }
  __syncthreads();
  if (wv == 0) {
#pragma unroll
    for (int r = 0; r < 8; r++) {
      float sn = 0.f, sd = 0.f;
      for (int w2 = 0; w2 < 8; w2++) { sn += redn[w2][r * 32 + lane]; sd += redd[w2][r * 32 + lane]; }
      if (hf == 0) {               // C layout: M = r (+8*half) = k, valid k<8 only
        ws[oPVn + ((size_t)yb * KB + r) * JD + j0 + l16] = sn;
        ws[oPVd + ((size_t)yb * KB + r) * JD + j0 + l16] = sd;
      }
    }
  }
}

// ---------------- NMF: V update stage 2 (deterministic partial sum + apply) ----------------
__global__ __launch_bounds__(256) void k_v_apply(float* __restrict__ ws, int n) {
  size_t t = (size_t)blockIdx.x * 256 + threadIdx.x;
  size_t tot = (size_t)KB * JD;
  if (t >= tot) return;
  float sn = 0.f, sd = 0.f;
#pragma unroll
  for (int y = 0; y < YCH; y++) {
    sn += ws[oPVn + (size_t)y * KB * JD + t];
    sd += ws[oPVd + (size_t)y * KB * JD + t];
  }
  ws[oV + (size_t)n * KB * JD + t] *= sqrtf(sn / (sd + EPS));
}

// ---------------- IP update: per frequency 2x2 weighted covariance + solve ----------------
__global__ __launch_bounds__(256) void k_ip_update(float* __restrict__ ws, int n) {
  const int i = blockIdx.x;      // < ID
  const int tid = threadIdx.x;
  const float* T = ws + oT + (size_t)n * IPD * KB;
  const float* V = ws + oV + (size_t)n * KB * JD;
  const float* X0r = ws + oXre;
  const float* X0i = ws + oXim;
  const float* X1r = ws + oXre + (size_t)IPD * JD;
  const float* X1i = ws + oXim + (size_t)IPD * JD;

  __shared__ float sT[KB];
  __shared__ float rb[4][256];
  if (tid < KB) sT[tid] = T[(size_t)i * KB + tid];
  __syncthreads();

  float d00 = 0.f, d11 = 0.f, d01r = 0.f, d01i = 0.f;
  for (int j = tid; j < JD; j += 256) {
    float rn = 0.f;
#pragma unroll
    for (int k = 0; k < KB; k++) rn += sT[k] * V[(size_t)k * JD + j];
    float s = sqrtf(1.0f / (rn + EPS));
    size_t idx = (size_t)i * JD + j;
    float x0r = X0r[idx] * s, x0i = X0i[idx] * s;
    float x1r = X1r[idx] * s, x1i = X1i[idx] * s;
    d00 += x0r * x0r + x0i * x0i;
    d11 += x1r * x1r + x1i * x1i;
    d01r += x0r * x1r + x0i * x1i;       // Xw0 * conj(Xw1)
    d01i += x0i * x1r - x0r * x1i;
  }
  rb[0][tid] = d00; rb[1][tid] = d11; rb[2][tid] = d01r; rb[3][tid] = d01i;
  __syncthreads();
  for (int s2 = 128; s2 > 0; s2 >>= 1) {
    if (tid < s2) {
      rb[0][tid] += rb[0][tid + s2]; rb[1][tid] += rb[1][tid + s2];
      rb[2][tid] += rb[2][tid + s2]; rb[3][tid] += rb[3][tid + s2];
    }
    __syncthreads();
  }
  if (tid == 0) {
    const float invJ = 1.0f / (float)JD;
    float D00 = rb[0][0] * invJ + EPS;   // real (Dreg diag)
    float D11 = rb[1][0] * invJ + EPS;
    float Dr = rb[2][0] * invJ, Di = rb[3][0] * invJ;  // D01 = Dr+iDi, D10 = conj
    float* Wf = ws + oW + (size_t)i * 8;
    float w00r = Wf[0], w00i = Wf[1], w01r = Wf[2], w01i = Wf[3];
    float w10r = Wf[4], w10i = Wf[5], w11r = Wf[6], w11i = Wf[7];
    // A = W @ Dreg
    float A00r = w00r * D00 + (w01r * Dr + w01i * Di);
    float A00i = w00i * D00 + (w01i * Dr - w01r * Di);
    float A01r = (w00r * Dr - w00i * Di) + w01r * D11;
    float A01i = (w00i * Dr + w00r * Di) + w01i * D11;
    float A10r = w10r * D00 + (w11r * Dr + w11i * Di);
    float A10i = w10i * D00 + (w11i * Dr - w11r * Di);
    float A11r = (w10r * Dr - w10i * Di) + w11r * D11;
    float A11i = (w10i * Dr + w10r * Di) + w11i * D11;
    // det = A00*A11 - A01*A10
    float detr = (A00r * A11r - A00i * A11i) - (A01r * A10r - A01i * A10i);
    float deti = (A00r * A11i + A00i * A11r) - (A01r * A10i + A01i * A10r);
    // b = column n of inv(A)
    float n0r, n0i, n1r, n1i;
    if (n == 0) { n0r = A11r;  n0i = A11i;  n1r = -A10r; n1i = -A10i; }
    else        { n0r = -A01r; n0i = -A01i; n1r = A00r;  n1i = A00i;  }
    float dd = detr * detr + deti * deti;
    float b0r = (n0r * detr + n0i * deti) / dd, b0i = (n0i * detr - n0r * deti) / dd;
    float b1r = (n1r * detr + n1i * deti) / dd, b1i = (n1i * detr - n1r * deti) / dd;
    // Db = Dreg @ b
    float Db0r = D00 * b0r + (Dr * b1r - Di * b1i);
    float Db0i = D00 * b0i + (Dr * b1i + Di * b1r);
    float Db1r = (Dr * b0r + Di * b0i) + D11 * b1r;
    float Db1i = (Dr * b0i - Di * b0r) + D11 * b1i;
    float denr = b0r * Db0r + b0i * Db0i + b1r * Db1r + b1i * Db1i;  // Re(b^H Db)
    float dnm = sqrtf(denr + EPS);
    // W[i,n,:] = conj(b / dnm)
    Wf[(n * 2 + 0) * 2]     = b0r / dnm;
    Wf[(n * 2 + 0) * 2 + 1] = -b0i / dnm;
    Wf[(n * 2 + 1) * 2]     = b1r / dnm;
    Wf[(n * 2 + 1) * 2 + 1] = -b1i / dnm;
  }
}

// ---------------- Y = W @ Xc (per frequency) ----------------
__global__ __launch_bounds__(256) void k_y_update(float* __restrict__ ws) {
  const int i = blockIdx.x;      // < ID
  const int tid = threadIdx.x;
  __shared__ float sw[8];
  if (tid < 8) sw[tid] = ws[oW + (size_t)i * 8 + tid];
  __syncthreads();
  const float* X0r = ws + oXre;
  const float* X0i = ws + oXim;
  const float* X1r = ws + oXre + (size_t)IPD * JD;
  const float* X1i = ws + oXim + (size_t)IPD * JD;
  float* Y0r = ws + oYre;
  float* Y0i = ws + oYim;
  float* Y1r = ws + oYre + (size_t)IPD * JD;
  float* Y1i = ws + oYim + (size_t)IPD * JD;
  for (int j = tid; j < JD; j += 256) {
    size_t idx = (size_t)i * JD + j;
    float x0r = X0r[idx], x0i = X0i[idx], x1r = X1r[idx], x1i = X1i[idx];
    {
      float wr0 = sw[0], wi0 = sw[1], wr1 = sw[2], wi1 = sw[3];
      Y0r[idx] = wr0 * x0r - wi0 * x0i + wr1 * x1r - wi1 * x1i;
      Y0i[idx] = wr0 * x0i + wi0 * x0r + wr1 * x1i + wi1 * x1r;
    }
    {
      float wr0 = sw[4], wi0 = sw[5], wr1 = sw[6], wi1 = sw[7];
      Y1r[idx] = wr0 * x0r - wi0 * x0i + wr1 * x1r - wi1 * x1i;
      Y1i[idx] = wr0 * x0i + wi0 * x0r + wr1 * x1i + wi1 * x1r;
    }
  }
}

// ---------------- output: (N, J, I, 2) ----------------
__global__ __launch_bounds__(256) void k_out(const float* __restrict__ ws, float* __restrict__ out) {
  size_t t = (size_t)blockIdx.x * 256 + threadIdx.x;
  size_t tot = (size_t)NSRC * JD * ID * 2;
  if (t >= tot) return;
  int c = (int)(t & 1); size_t r = t >> 1;
  int i = (int)(r % ID); r /= ID;
  int j = (int)(r % JD); int n = (int)(r / JD);
  size_t idx = ((size_t)n * IPD + i) * JD + j;
  out[t] = (c == 0) ? ws[oYre + idx] : ws[oYim + idx];
}

extern "C" void kernel_launch(void* const* d_in, const int* in_sizes, int n_in,
                              void* d_out, int out_size, void* d_ws, size_t ws_size,
                              hipStream_t stream) {
  (void)in_sizes; (void)n_in; (void)out_size; (void)ws_size;
  const float* X  = (const float*)d_in[0];
  const float* T0 = (const float*)d_in[1];
  const float* V0 = (const float*)d_in[2];
  float* ws = (float*)d_ws;
  float* out = (float*)d_out;

  { size_t tot = (size_t)MC * IPD * JD;   k_init_xy<<<(unsigned)((tot + 255) / 256), 256, 0, stream>>>(X, ws); }
  { size_t tot = (size_t)NSRC * IPD * KB; k_init_t<<<(unsigned)((tot + 255) / 256), 256, 0, stream>>>(T0, ws); }
  { size_t tot = (size_t)NSRC * KB * JD;  k_init_v<<<(unsigned)((tot + 255) / 256), 256, 0, stream>>>(V0, ws); }
  { size_t tot = (size_t)ID * 8;          k_init_w<<<(unsigned)((tot + 255) / 256), 256, 0, stream>>>(ws); }

  for (int it = 0; it < NITER; ++it) {
    for (int n = 0; n < NSRC; ++n) {
      k_t_update<<<dim3(IPD / 16, YCH), 256, 0, stream>>>(ws, n);
      k_t_apply<<<(unsigned)(((size_t)IPD * KB + 255) / 256), 256, 0, stream>>>(ws, n);
      k_v_update<<<dim3(JD / 16, YCH), 256, 0, stream>>>(ws, n);
      k_v_apply<<<(unsigned)(((size_t)KB * JD + 255) / 256), 256, 0, stream>>>(ws, n);
      k_ip_update<<<ID, 256, 0, stream>>>(ws, n);
    }
    k_y_update<<<ID, 256, 0, stream>>>(ws);
  }
  { size_t tot = (size_t)NSRC * JD * ID * 2; k_out<<<(unsigned)((tot + 255) / 256), 256, 0, stream>>>(ws, out); }
}